// SimpleNet_67319317397983
// MI455X (gfx1250) — compile-verified
//
#include <hip/hip_runtime.h>
#include <hip/hip_bf16.h>
#include <stdint.h>

// rel_pos gather: out[j, h] = W[h, rel_pos[j]],  j over 4*2048*2048, h in [0,12)
// Pure-bandwidth op: ~872 MB traffic -> ~37 us floor at 23.3 TB/s.
// CDNA5 path: TDM tensor_load_to_lds broadcasts W (3 KB) into LDS per block.

typedef unsigned int u32x4 __attribute__((ext_vector_type(4)));
typedef int          i32x4 __attribute__((ext_vector_type(4)));
typedef int          i32x8 __attribute__((ext_vector_type(8)));
typedef float        f32x4 __attribute__((ext_vector_type(4)));  // native clang vector (NT-store OK)

#define NUM_HEADS    12
#define NUM_CLASSES  64
#define TABLE_ELTS   (NUM_HEADS * NUM_CLASSES)   // 768 f32 = 3072 bytes

__global__ __launch_bounds__(256) void relpos_gather_kernel(
    const int* __restrict__ rel_pos,   // [16M] values in [0,64)
    const float* __restrict__ W,       // [12,64] row-major
    float* __restrict__ out,           // [16M * 12] f32
    unsigned n4)                       // out elements / 4
{
    __shared__ float w_lds[TABLE_ELTS];

    // ---- TDM: async DMA of W (12x64 f32) global -> LDS, wave 0 only ----
    if (threadIdx.x < 32u) {
        unsigned long long ga = (unsigned long long)(const void*)W;
        unsigned lds_off = (unsigned)(unsigned long long)(void*)&w_lds[0];

        // D# group 0 (128b): count=1 | lds_addr | global_addr[56:0] | type=2
        u32x4 g0;
        g0.x = 1u;                                     // count=1, user mode
        g0.y = lds_off;                                // lds_addr (bytes)
        g0.z = (unsigned)(ga & 0xFFFFFFFFull);         // global_addr[31:0]
        g0.w = (unsigned)((ga >> 32) & 0x01FFFFFFull)  // global_addr[56:32]
             | (2u << 30);                             // type = 2 ("image")

        // D# group 1 (256b): 1-row tile, 768 x 4B elements
        i32x8 g1;
        g1[0] = (int)(2u << 16);                       // data_size=2 (4B), wg_mask=0
        g1[1] = (int)((unsigned)TABLE_ELTS << 16);     // tensor_dim0[15:0] @ [63:48]
        g1[2] = (int)(1u << 16);                       // tensor_dim0 hi=0; tensor_dim1=1 @ [95:80]
        g1[3] = (int)((unsigned)TABLE_ELTS << 16);     // tile_dim0=768 @ [127:112]
        g1[4] = 1;                                     // tile_dim1=1, tile_dim2=0
        g1[5] = TABLE_ELTS;                            // tensor_dim0_stride[31:0]
        g1[6] = (int)((unsigned)TABLE_ELTS << 16);     // tensor_dim1_stride[15:0] @ [223:208]
        g1[7] = 0;

        i32x4 gz = {0, 0, 0, 0};                       // groups 2/3 unused (2D tensor)

#if defined(__clang_major__) && (__clang_major__ >= 23)
        i32x8 gz8 = {0, 0, 0, 0, 0, 0, 0, 0};
        __builtin_amdgcn_tensor_load_to_lds(g0, g1, gz, gz, gz8, 0);
#else
        __builtin_amdgcn_tensor_load_to_lds(g0, g1, gz, gz, 0);
#endif
        __builtin_amdgcn_s_wait_tensorcnt(0);          // s_wait_tensorcnt 0
    }
    __syncthreads();

    const unsigned stride = gridDim.x * blockDim.x;
    f32x4* __restrict__ out4 = (f32x4*)out;

    for (unsigned i = blockIdx.x * blockDim.x + threadIdx.x; i < n4; i += stride) {
        unsigned e0 = i * 4u;                 // first flat output element
        unsigned j  = e0 / 12u;               // rel_pos index (mulhi, no real div)
        unsigned c  = e0 - j * 12u;           // head channel within [0,12)
        unsigned jl = (e0 + 3u) / 12u;        // last rel_pos index touched (<= j+1)

        int idx  = rel_pos[j];
        int idxl = rel_pos[jl];               // same address when jl==j (cache hit)

        f32x4 v;
#pragma unroll
        for (int k = 0; k < 4; ++k) {
            // LDS bank == idx (idx < 64): distinct indices across lanes conflict-free
            v[k] = w_lds[c * 64u + (unsigned)idx];
            ++c;
            if (c == 12u) { c = 0u; idx = idxl; }
        }
        // 805 MB streamed once, never re-read: bypass temporal caching.
        __builtin_nontemporal_store(v, &out4[i]);
    }
}

extern "C" void kernel_launch(void* const* d_in, const int* in_sizes, int n_in,
                              void* d_out, int out_size, void* d_ws, size_t ws_size,
                              hipStream_t stream) {
    const int*   rel_pos = (const int*)d_in[0];     // [4,2048,2048] int32
    // d_in[1] = hidden_states: unused by the reference math (dtype carrier only)
    const float* W       = (const float*)d_in[2];   // [12,64] f32
    float* out = (float*)d_out;                     // [4,2048,2048,12] f32

    unsigned n4 = (unsigned)(out_size / 4);         // out_size divisible by 4 (12*4M)
    const int threads = 256;                        // 8 wave32 per block
    const int blocks  = 4096;                       // persistent: ~4K TDM ops total

    relpos_gather_kernel<<<blocks, threads, 0, stream>>>(rel_pos, W, out, n4);
}